// VectorQuantizer_43233140802031
// MI455X (gfx1250) — compile-verified
//
#include <hip/hip_runtime.h>

typedef float v2f __attribute__((ext_vector_type(2)));
typedef float v4f __attribute__((ext_vector_type(4)));
typedef float v8f __attribute__((ext_vector_type(8)));

#define NUM_E 1024      // codebook entries (K)
#define DIM   64        // embedding dim (D)
#define NPTS  32768     // 32 * 32 * 32 points (N)
#define XB_STRIDE 65536 // 64*1024 elements per batch image in x [B,D,H,W]

// ---------------------------------------------------------------------------
// Kernel 1: c2[k] = 0.5 * ||codebook[k]||^2   (argmin(d2) == argmin(c2 - x.c))
// ---------------------------------------------------------------------------
__global__ void vq_c2_kernel(const float* __restrict__ cb, float* __restrict__ c2) {
  int k = blockIdx.x * blockDim.x + threadIdx.x;
  if (k >= NUM_E) return;
  const v4f* row = (const v4f*)(cb + (size_t)k * DIM);
  float s = 0.f;
#pragma unroll
  for (int i = 0; i < DIM / 4; ++i) {
    v4f v = row[i];
    s += v.x * v.x + v.y * v.y + v.z * v.z + v.w * v.w;
  }
  c2[k] = 0.5f * s;
}

// ---------------------------------------------------------------------------
// Kernel 2: fused WMMA distance + argmin + one-hot + gather.
// One wave = 16 points x full 1024-entry codebook.
// ---------------------------------------------------------------------------
__global__ void __launch_bounds__(256)
vq_main_kernel(const float* __restrict__ x, const float* __restrict__ cb,
               const float* __restrict__ c2, float* __restrict__ enc,
               float* __restrict__ qout) {
  const int lane   = threadIdx.x & 31;
  const int waveId = threadIdx.x >> 5;
  const int lr     = lane & 15;   // lane-in-half
  const int half   = lane >> 4;   // 0: K0/K1 (and M 0-7 of C), 1: K2/K3 (M 8-15)
  const int n0     = (blockIdx.x * 8 + waveId) * 16;

  // ---- A fragments: 16 rows x 64 dims, reused across all 64 column tiles ---
  // 32-bit 16x4 A layout: lanes 0-15 hold rows; VGPR0=K0(+2*half), VGPR1=K1(+2*half)
  const int rowA = n0 + lr;
  const int bA   = rowA >> 10;
  const int hwA  = rowA & 1023;
  const size_t xbase = (size_t)bA * XB_STRIDE + hwA;   // x[b, d, h, w] d-stride 1024
  v2f a[16];
#pragma unroll
  for (int q = 0; q < 16; ++q) {
    const int d0 = 4 * q + 2 * half;
    a[q].x = x[xbase + (size_t)d0 * 1024];
    a[q].y = x[xbase + (size_t)(d0 + 1) * 1024];
  }

  float bestv[8];
  int   besti[8];
#pragma unroll
  for (int r = 0; r < 8; ++r) { bestv[r] = 3.4e38f; besti[r] = 0; }

  // ---- sweep the 64 column tiles of the codebook -----------------------
  for (int c = 0; c < NUM_E / 16; ++c) {
    const int col = c * 16 + lr;
    // B (4x16) mirrors A with N instead of M: lane<16 -> K0,K1 of column `col`
    const v2f* cbrow = (const v2f*)(cb + (size_t)col * DIM);
    v8f acc = {};
#pragma unroll
    for (int q = 0; q < 16; ++q) {
      const v2f bb = cbrow[2 * q + half];
      acc = __builtin_amdgcn_wmma_f32_16x16x4_f32(
          /*neg_a=*/false, a[q], /*neg_b=*/false, bb,
          /*c_mod=*/(short)0, acc, /*reuse_a=*/false, /*reuse_b=*/false);
    }
    const float c2v = c2[col];
#pragma unroll
    for (int r = 0; r < 8; ++r) {
      const float score = c2v - acc[r];   // 0.5||c||^2 - x.c
      if (score < bestv[r]) { bestv[r] = score; besti[r] = col; }
    }
  }

  // ---- argmin reduce across the 16 lanes of each half-wave --------------
  // C/D layout: lane L, VGPR r holds (M = r + 8*half, N = L%16); columns of one
  // row live across lr=0..15 at fixed r. xor offsets 1,2,4,8 stay within a half.
#pragma unroll
  for (int r = 0; r < 8; ++r) {
#pragma unroll
    for (int off = 8; off > 0; off >>= 1) {
      const float ov = __shfl_xor(bestv[r], off, 32);
      const int   oi = __shfl_xor(besti[r], off, 32);
      if (ov < bestv[r] || (ov == bestv[r] && oi < besti[r])) {
        bestv[r] = ov; besti[r] = oi;
      }
    }
  }

  // ---- outputs: half h owns rows n0+8h .. n0+8h+7 -----------------------
#pragma unroll
  for (int r = 0; r < 8; ++r) {
    const int row = n0 + half * 8 + r;
    const int idx = besti[r];
    const int b   = row >> 10;
    const int hw  = row & 1023;

    // quantized [B,D,H,W]: 16 lanes x 4 dims each, d-stride 1024
    const v4f cv = ((const v4f*)(cb + (size_t)idx * DIM))[lr];
    const size_t qbase = (size_t)b * XB_STRIDE + hw;
    qout[qbase + (size_t)(4 * lr + 0) * 1024] = cv.x;
    qout[qbase + (size_t)(4 * lr + 1) * 1024] = cv.y;
    qout[qbase + (size_t)(4 * lr + 2) * 1024] = cv.z;
    qout[qbase + (size_t)(4 * lr + 3) * 1024] = cv.w;

    // one-hot encodings row: zeros+one fused, each of 1024 floats written once;
    // nontemporal so the 128 MB output stream doesn't thrash L2.
    float* erow = enc + (size_t)row * NUM_E;
#pragma unroll
    for (int p = 0; p < 16; ++p) {
      const int base = p * 64 + 4 * lr;
      v4f ov;
      ov.x = (base + 0 == idx) ? 1.f : 0.f;
      ov.y = (base + 1 == idx) ? 1.f : 0.f;
      ov.z = (base + 2 == idx) ? 1.f : 0.f;
      ov.w = (base + 3 == idx) ? 1.f : 0.f;
      __builtin_nontemporal_store(ov, (v4f*)(erow + base));
    }
  }
}

extern "C" void kernel_launch(void* const* d_in, const int* in_sizes, int n_in,
                              void* d_out, int out_size, void* d_ws, size_t ws_size,
                              hipStream_t stream) {
  const float* x  = (const float*)d_in[0];   // [32,64,32,32] f32
  const float* cb = (const float*)d_in[1];   // [1024,64] f32
  float* enc  = (float*)d_out;                        // [32768,1024]
  float* qout = enc + (size_t)NPTS * NUM_E;           // [32,64,32,32]
  float* c2   = (float*)d_ws;                         // [1024] scratch

  vq_c2_kernel<<<NUM_E / 256, 256, 0, stream>>>(cb, c2);
  vq_main_kernel<<<NPTS / 128, 256, 0, stream>>>(x, cb, c2, enc, qout);
}